// data_consistency_gmres_83734682403049
// MI455X (gfx1250) — compile-verified
//
#include <hip/hip_runtime.h>
#include <hip/hip_bf16.h>
#include <math.h>

// ---------------- problem constants ----------------
constexpr int Bq   = 4;
constexpr int Cc   = 16;
constexpr int N    = 384;           // H == W == 384
constexpr int N2   = N * N;         // 147456 (also H*W)
constexpr int NV   = Bq * N2;       // 589824  elements per Krylov plane
constexpr int NC   = Bq * Cc * N2;  // 9437184 elements per coil plane
constexpr int MIT  = 8;             // GMRES dimension
constexpr float LAM0 = 0.05f;

// 32x32 output region per wave: (384/32)^2 = 144 wave-tiles per image
constexpr int WTILES = 12 * 12;     // 144
constexpr int NBATCH = Bq * Cc;     // 64 images per GEMM

// ---------------- workspace layout (floats) ----------------
constexpr size_t OFF_FR   = 0;
constexpr size_t OFF_FI   = OFF_FR + N2;
constexpr size_t OFF_FIN  = OFF_FI + N2;
constexpr size_t OFF_Q    = OFF_FIN + N2;                 // 9 vectors x (2*NV)
constexpr size_t OFF_W    = OFF_Q + (size_t)9 * 2 * NV;   // 2*NV
constexpr size_t OFF_B1   = OFF_W + (size_t)2 * NV;       // 2*NC
constexpr size_t OFF_B2   = OFF_B1 + (size_t)2 * NC;      // 2*NC
constexpr size_t OFF_PART = OFF_B2 + (size_t)2 * NC;      // 1024
constexpr size_t OFF_SC   = OFF_PART + 1024;              // 256 scalars

// scalar slots (relative to OFF_SC)
constexpr int SC_BETA = 0;
constexpr int SC_INV  = 1;
constexpr int SC_H    = 2;            // 9x8 complex: SC_H + (j*8+k)*2
constexpr int SC_Y    = SC_H + 144;   // 8 complex

typedef float v2f __attribute__((ext_vector_type(2)));
typedef float v8f __attribute__((ext_vector_type(8)));

// ---------------- DFT matrix init: F = exp(-2*pi*i*j*k/N) ----------------
__global__ void dft_init_kernel(float* __restrict__ Fr, float* __restrict__ Fi,
                                float* __restrict__ FiN) {
  int i = blockIdx.x * blockDim.x + threadIdx.x;
  if (i >= N2) return;
  int j = i / N, k = i % N;
  int p = (j * k) % N;
  float ang = -2.0f * 3.14159265358979323846f * (float)p / (float)N;
  float sn, cn;
  __sincosf(ang, &sn, &cn);
  Fr[i] = cn;          // real part
  Fi[i] = sn;          // imag part (sin of negative angle)
  FiN[i] = -sn;        // conj plane for inverse transform
}

// ---------------- batched 384x384 complex GEMM via V_WMMA_F32_16X16X4_F32 --
// out[b] = A[b?] * B[b?]  (stride==0 -> shared operand, e.g. DFT matrix)
// Re = Ar*Br - Ai*Bi ; Im = Ar*Bi + Ai*Br, 4 separate WMMA chains per tile.
// One wave computes a 2x2 block of 16x16 tiles (32x32 output): A fragments
// are reused across the 2 column tiles and B fragments across the 2 row
// tiles, halving cache traffic per WMMA.
__global__ void __launch_bounds__(128)
cgemm384_kernel(float* __restrict__ outR, float* __restrict__ outI,
                const float* __restrict__ aR, const float* __restrict__ aI,
                long long aStride,
                const float* __restrict__ bR, const float* __restrict__ bI,
                long long bStride,
                const int* __restrict__ mask, float scale) {
  const int wave = blockIdx.x * 4 + (threadIdx.x >> 5);
  const int lane = threadIdx.x & 31;
  const int batch = wave / WTILES;
  const int tile  = wave % WTILES;
  const int tr = tile / 12, tc = tile % 12;   // 32x32 block coords

  const float* Ar = aR + (long long)batch * aStride;
  const float* Ai = aI + (long long)batch * aStride;
  const float* Br = bR + (long long)batch * bStride;
  const float* Bi = bI + (long long)batch * bStride;

  const int halfSel = lane >> 4;   // 0: lanes 0-15, 1: lanes 16-31
  const int l15 = lane & 15;
  const int arow = tr * 32 + l15;  // A fragment row (tile mi adds +16)
  const int bcol = tc * 32 + l15;  // B/C/D fragment col (tile ni adds +16)

  // A 16x4 layout: vgpr0 = K{0|2}, vgpr1 = K{1|3} -> contiguous pair per lane.
  const float* pAr = Ar + (long long)arow * N + 2 * halfSel;
  const float* pAi = Ai + (long long)arow * N + 2 * halfSel;
  // B 4x16 layout: vgpr v holds row K = v + 2*halfSel.
  const float* pBr = Br + (long long)(2 * halfSel) * N + bcol;
  const float* pBi = Bi + (long long)(2 * halfSel) * N + bcol;

  v8f accRR[2][2] = {}; v8f accII[2][2] = {};
  v8f accRI[2][2] = {}; v8f accIR[2][2] = {};

  for (int kk = 0; kk < N; kk += 4) {
    v2f arf[2], aif[2], brf[2], bif[2];
    arf[0] = *(const v2f*)(pAr);
    arf[1] = *(const v2f*)(pAr + 16 * N);
    aif[0] = *(const v2f*)(pAi);
    aif[1] = *(const v2f*)(pAi + 16 * N);
    brf[0].x = pBr[0];  brf[0].y = pBr[N];
    brf[1].x = pBr[16]; brf[1].y = pBr[N + 16];
    bif[0].x = pBi[0];  bif[0].y = pBi[N];
    bif[1].x = pBi[16]; bif[1].y = pBi[N + 16];
#pragma unroll
    for (int mi = 0; mi < 2; mi++) {
#pragma unroll
      for (int ni = 0; ni < 2; ni++) {
        accRR[mi][ni] = __builtin_amdgcn_wmma_f32_16x16x4_f32(
            false, arf[mi], false, brf[ni], (short)0, accRR[mi][ni], false, false);
        accII[mi][ni] = __builtin_amdgcn_wmma_f32_16x16x4_f32(
            false, aif[mi], false, bif[ni], (short)0, accII[mi][ni], false, false);
        accRI[mi][ni] = __builtin_amdgcn_wmma_f32_16x16x4_f32(
            false, arf[mi], false, bif[ni], (short)0, accRI[mi][ni], false, false);
        accIR[mi][ni] = __builtin_amdgcn_wmma_f32_16x16x4_f32(
            false, aif[mi], false, brf[ni], (short)0, accIR[mi][ni], false, false);
      }
    }
    pAr += 4; pAi += 4; pBr += 4 * N; pBi += 4 * N;
  }

  const long long obase = (long long)batch * N2;
#pragma unroll
  for (int mi = 0; mi < 2; mi++) {
#pragma unroll
    for (int ni = 0; ni < 2; ni++) {
      int ocol = tc * 32 + ni * 16 + l15;
#pragma unroll
      for (int v = 0; v < 8; v++) {
        int orow = tr * 32 + mi * 16 + v + halfSel * 8;  // C/D: vgpr v -> M=v (+8 upper)
        long long oidx = obase + (long long)orow * N + ocol;
        float re = (accRR[mi][ni][v] - accII[mi][ni][v]) * scale;
        float im = (accRI[mi][ni][v] + accIR[mi][ni][v]) * scale;
        if (mask) {
          float m = (float)mask[orow * N + ocol];
          re *= m; im *= m;
        }
        outR[oidx] = re;
        outI[oidx] = im;
      }
    }
  }
}

// ---------------- elementwise kernels ----------------
__global__ void rhs_kernel(const float* __restrict__ x0r, const float* __restrict__ x0i,
                           const float* __restrict__ zr, const float* __restrict__ zi,
                           const float* __restrict__ lam,
                           float* __restrict__ wr, float* __restrict__ wi) {
  int i = blockIdx.x * blockDim.x + threadIdx.x;
  if (i >= NV) return;
  float ls = lam[0] * LAM0;
  wr[i] = x0r[i] + ls * zr[i];
  wi[i] = x0i[i] + ls * zi[i];
}

// coil = csm * x   (broadcast x over coils)
__global__ void coil_expand_kernel(const float* __restrict__ cr, const float* __restrict__ ci,
                                   const float* __restrict__ xr, const float* __restrict__ xi,
                                   float* __restrict__ or_, float* __restrict__ oi_) {
  int i = blockIdx.x * blockDim.x + threadIdx.x;
  if (i >= NC) return;
  int pix = i % N2;
  int b = i / (Cc * N2);
  int xidx = b * N2 + pix;
  float c_r = cr[i], c_i = ci[i], x_r = xr[xidx], x_i = xi[xidx];
  or_[i] = c_r * x_r - c_i * x_i;
  oi_[i] = c_r * x_i + c_i * x_r;
}

// w = sum_c conj(csm) * g  +  lam_s * x
__global__ void coil_combine_kernel(const float* __restrict__ cr, const float* __restrict__ ci,
                                    const float* __restrict__ gr, const float* __restrict__ gi,
                                    const float* __restrict__ xr, const float* __restrict__ xi,
                                    const float* __restrict__ lam,
                                    float* __restrict__ wr, float* __restrict__ wi) {
  int i = blockIdx.x * blockDim.x + threadIdx.x;
  if (i >= NV) return;
  int pix = i % N2;
  int b = i / N2;
  float accr = 0.f, acci = 0.f;
#pragma unroll 4
  for (int c = 0; c < Cc; c++) {
    int idx = (b * Cc + c) * N2 + pix;
    float c_r = cr[idx], c_i = ci[idx], g_r = gr[idx], g_i = gi[idx];
    accr += c_r * g_r + c_i * g_i;        // Re(conj(c)*g)
    acci += c_r * g_i - c_i * g_r;        // Im(conj(c)*g)
  }
  float ls = lam[0] * LAM0;
  wr[i] = accr + ls * xr[i];
  wi[i] = acci + ls * xi[i];
}

// partial <conj(a), b>, fixed grid -> deterministic
__global__ void dot_kernel(const float* __restrict__ ar, const float* __restrict__ ai,
                           const float* __restrict__ br, const float* __restrict__ bi,
                           float* __restrict__ partials) {
  __shared__ float sr[256], si[256];
  float re = 0.f, im = 0.f;
  for (int i = blockIdx.x * blockDim.x + threadIdx.x; i < NV;
       i += gridDim.x * blockDim.x) {
    float a_r = ar[i], a_i = ai[i], b_r = br[i], b_i = bi[i];
    re += a_r * b_r + a_i * b_i;
    im += a_r * b_i - a_i * b_r;
  }
  sr[threadIdx.x] = re; si[threadIdx.x] = im;
  __syncthreads();
  for (int s = 128; s > 0; s >>= 1) {
    if (threadIdx.x < (unsigned)s) {
      sr[threadIdx.x] += sr[threadIdx.x + s];
      si[threadIdx.x] += si[threadIdx.x + s];
    }
    __syncthreads();
  }
  if (threadIdx.x == 0) {
    partials[blockIdx.x * 2 + 0] = sr[0];
    partials[blockIdx.x * 2 + 1] = si[0];
  }
}

// mode 0: complex dot -> dst[0..1]
// mode 1: norm -> dst[0]=sqrt(re), dst[1]=0, inv[0]= hn>0 ? 1/hn : 0
// mode 2: beta -> dst[0]=sqrt(re), dst[1]=0, inv[0]= 1/beta
__global__ void finalize_kernel(const float* __restrict__ partials, int nblocks,
                                float* __restrict__ dst, float* __restrict__ inv,
                                int mode) {
  if (threadIdx.x != 0 || blockIdx.x != 0) return;
  float re = 0.f, im = 0.f;
  for (int i = 0; i < nblocks; i++) {
    re += partials[2 * i];
    im += partials[2 * i + 1];
  }
  if (mode == 0) {
    dst[0] = re; dst[1] = im;
  } else {
    float hn = sqrtf(re);
    dst[0] = hn; dst[1] = 0.f;
    if (mode == 1) inv[0] = (hn > 0.f) ? 1.f / hn : 0.f;
    else           inv[0] = 1.f / hn;
  }
}

// w -= h * q   (complex)
__global__ void axpy_kernel(float* __restrict__ wr, float* __restrict__ wi,
                            const float* __restrict__ qr, const float* __restrict__ qi,
                            const float* __restrict__ h) {
  int i = blockIdx.x * blockDim.x + threadIdx.x;
  if (i >= NV) return;
  float hr = h[0], hi = h[1];
  float q_r = qr[i], q_i = qi[i];
  wr[i] -= hr * q_r - hi * q_i;
  wi[i] -= hr * q_i + hi * q_r;
}

// dst = w * inv[0]
__global__ void scale_kernel(const float* __restrict__ wr, const float* __restrict__ wi,
                             const float* __restrict__ inv,
                             float* __restrict__ dr, float* __restrict__ di) {
  int i = blockIdx.x * blockDim.x + threadIdx.x;
  if (i >= NV) return;
  float s = inv[0];
  dr[i] = wr[i] * s;
  di[i] = wi[i] * s;
}

// ---------------- tiny complex helpers + Givens QR solve (1 thread) -------
struct cpx { float r, i; };
__device__ inline cpx cadd(cpx a, cpx b) { return {a.r + b.r, a.i + b.i}; }
__device__ inline cpx csub(cpx a, cpx b) { return {a.r - b.r, a.i - b.i}; }
__device__ inline cpx cmul(cpx a, cpx b) { return {a.r * b.r - a.i * b.i, a.r * b.i + a.i * b.r}; }
__device__ inline cpx cconj(cpx a) { return {a.r, -a.i}; }
__device__ inline cpx cscale(cpx a, float s) { return {a.r * s, a.i * s}; }
__device__ inline float cabs2(cpx a) { return a.r * a.r + a.i * a.i; }
__device__ inline cpx cdiv(cpx a, cpx b) {
  float d = cabs2(b);
  float id = (d > 0.f) ? 1.f / d : 0.f;
  return cscale(cmul(a, cconj(b)), id);
}

__global__ void gmres_solve_kernel(float* __restrict__ sc) {
  if (threadIdx.x != 0 || blockIdx.x != 0) return;
  cpx Hm[MIT + 1][MIT];
  for (int k = 0; k < MIT; k++)
    for (int j = 0; j <= MIT; j++) {
      if (j <= k + 1) {
        Hm[j][k].r = sc[SC_H + (j * MIT + k) * 2 + 0];
        Hm[j][k].i = sc[SC_H + (j * MIT + k) * 2 + 1];
      } else {
        Hm[j][k] = {0.f, 0.f};
      }
    }
  cpx g[MIT + 1];
  g[0] = {sc[SC_BETA], 0.f};
  for (int j = 1; j <= MIT; j++) g[j] = {0.f, 0.f};

  float cs[MIT]; cpx sn[MIT];
  for (int k = 0; k < MIT; k++) {
    for (int j = 0; j < k; j++) {
      cpx t0 = Hm[j][k], t1 = Hm[j + 1][k];
      Hm[j][k]     = cadd(cscale(t0, cs[j]), cmul(sn[j], t1));
      Hm[j + 1][k] = csub(cscale(t1, cs[j]), cmul(cconj(sn[j]), t0));
    }
    cpx a = Hm[k][k], b = Hm[k + 1][k];
    float t = sqrtf(cabs2(a) + cabs2(b));
    float c; cpx s;
    if (t <= 0.f) {
      c = 1.f; s = {0.f, 0.f};
    } else {
      float aa = sqrtf(cabs2(a));
      if (aa <= 0.f) {
        c = 0.f; s = cscale(cconj(b), 1.f / t);
      } else {
        c = aa / t;
        cpx sgn = cscale(a, 1.f / aa);
        s = cscale(cmul(sgn, cconj(b)), 1.f / t);
      }
    }
    cs[k] = c; sn[k] = s;
    Hm[k][k] = cadd(cscale(a, c), cmul(s, b));
    Hm[k + 1][k] = {0.f, 0.f};
    cpx g0 = g[k], g1 = g[k + 1];
    g[k]     = cadd(cscale(g0, c), cmul(s, g1));
    g[k + 1] = csub(cscale(g1, c), cmul(cconj(s), g0));
  }
  cpx y[MIT];
  for (int k = MIT - 1; k >= 0; k--) {
    cpx acc = g[k];
    for (int j = k + 1; j < MIT; j++) acc = csub(acc, cmul(Hm[k][j], y[j]));
    y[k] = cdiv(acc, Hm[k][k]);
    sc[SC_Y + 2 * k + 0] = y[k].r;
    sc[SC_Y + 2 * k + 1] = y[k].i;
  }
}

// out[0,:]=Re(sum y_k Q_k), out[1,:]=Im(...)
__global__ void combine_out_kernel(const float* __restrict__ qbase,
                                   const float* __restrict__ sc,
                                   float* __restrict__ out) {
  int i = blockIdx.x * blockDim.x + threadIdx.x;
  if (i >= NV) return;
  float accr = 0.f, acci = 0.f;
#pragma unroll
  for (int k = 0; k < MIT; k++) {
    float yr = sc[SC_Y + 2 * k], yi = sc[SC_Y + 2 * k + 1];
    const float* qr = qbase + (size_t)k * (2 * NV);
    const float* qi = qr + NV;
    float q_r = qr[i], q_i = qi[i];
    accr += yr * q_r - yi * q_i;
    acci += yr * q_i + yi * q_r;
  }
  out[i] = accr;
  out[NV + i] = acci;
}

// ---------------- host orchestration ----------------
extern "C" void kernel_launch(void* const* d_in, const int* in_sizes, int n_in,
                              void* d_out, int out_size, void* d_ws, size_t ws_size,
                              hipStream_t stream) {
  const float* zkr  = (const float*)d_in[0];
  const float* zki  = (const float*)d_in[1];
  const float* x0r  = (const float*)d_in[2];
  const float* x0i  = (const float*)d_in[3];
  const float* csmr = (const float*)d_in[4];
  const float* csmi = (const float*)d_in[5];
  const int*   mask = (const int*)d_in[6];
  const float* lam  = (const float*)d_in[7];
  float* out = (float*)d_out;

  float* ws = (float*)d_ws;
  float* Fr  = ws + OFF_FR;
  float* Fi  = ws + OFF_FI;
  float* FiN = ws + OFF_FIN;
  float* wr  = ws + OFF_W;
  float* wi  = wr + NV;
  float* b1r = ws + OFF_B1;
  float* b1i = b1r + NC;
  float* b2r = ws + OFF_B2;
  float* b2i = b2r + NC;
  float* part = ws + OFF_PART;
  float* sc   = ws + OFF_SC;

  auto Qr = [&](int k) { return ws + OFF_Q + (size_t)k * 2 * NV; };
  auto Qi = [&](int k) { return ws + OFF_Q + (size_t)k * 2 * NV + NV; };

  const int gv = (NV + 255) / 256;        // vector-sized elementwise grids
  const int gc = (NC + 255) / 256;        // coil-sized elementwise grids
  const int gg = (NBATCH * WTILES) / 4;   // 2304 blocks, 4 waves each
  const float s = 1.0f / sqrtf((float)N); // ortho-norm per 1D transform

  // DFT matrix planes (recomputed each call: no cached state)
  dft_init_kernel<<<(N2 + 255) / 256, 256, 0, stream>>>(Fr, Fi, FiN);

  // rhs = x0 + lam_s * z_k  -> w ; beta = ||w|| ; Q0 = w / beta
  rhs_kernel<<<gv, 256, 0, stream>>>(x0r, x0i, zkr, zki, lam, wr, wi);
  dot_kernel<<<512, 256, 0, stream>>>(wr, wi, wr, wi, part);
  finalize_kernel<<<1, 32, 0, stream>>>(part, 512, sc + SC_BETA, sc + SC_INV, 2);
  scale_kernel<<<gv, 256, 0, stream>>>(wr, wi, sc + SC_INV, Qr(0), Qi(0));

  for (int k = 0; k < MIT; k++) {
    // ---- w = AtA(Q_k) + lam_s * Q_k ----
    coil_expand_kernel<<<gc, 256, 0, stream>>>(csmr, csmi, Qr(k), Qi(k), b1r, b1i);
    // fft2 = F * X * F (ortho): row transform then col transform (+mask)
    cgemm384_kernel<<<gg, 128, 0, stream>>>(b2r, b2i, b1r, b1i, (long long)N2,
                                            Fr, Fi, 0LL, nullptr, s);
    cgemm384_kernel<<<gg, 128, 0, stream>>>(b1r, b1i, Fr, Fi, 0LL,
                                            b2r, b2i, (long long)N2, mask, s);
    // ifft2 = conj(F) * Y * conj(F) (ortho)
    cgemm384_kernel<<<gg, 128, 0, stream>>>(b2r, b2i, b1r, b1i, (long long)N2,
                                            Fr, FiN, 0LL, nullptr, s);
    cgemm384_kernel<<<gg, 128, 0, stream>>>(b1r, b1i, Fr, FiN, 0LL,
                                            b2r, b2i, (long long)N2, nullptr, s);
    coil_combine_kernel<<<gv, 256, 0, stream>>>(csmr, csmi, b1r, b1i,
                                                Qr(k), Qi(k), lam, wr, wi);
    // ---- Gram-Schmidt (sequential, matches reference) ----
    for (int j = 0; j <= k; j++) {
      float* hslot = sc + SC_H + (j * MIT + k) * 2;
      dot_kernel<<<512, 256, 0, stream>>>(Qr(j), Qi(j), wr, wi, part);
      finalize_kernel<<<1, 32, 0, stream>>>(part, 512, hslot, nullptr, 0);
      axpy_kernel<<<gv, 256, 0, stream>>>(wr, wi, Qr(j), Qi(j), hslot);
    }
    // ---- h[k+1,k] = ||w||, Q_{k+1} = w/hn (0 if hn==0) ----
    float* hn_slot = sc + SC_H + ((k + 1) * MIT + k) * 2;
    dot_kernel<<<512, 256, 0, stream>>>(wr, wi, wr, wi, part);
    finalize_kernel<<<1, 32, 0, stream>>>(part, 512, hn_slot, sc + SC_INV, 1);
    scale_kernel<<<gv, 256, 0, stream>>>(wr, wi, sc + SC_INV, Qr(k + 1), Qi(k + 1));
  }

  // small least-squares + combine
  gmres_solve_kernel<<<1, 1, 0, stream>>>(sc);
  combine_out_kernel<<<gv, 256, 0, stream>>>(ws + OFF_Q, sc, out);
}